// BasicDeconvolutionBlock_52209622450767
// MI455X (gfx1250) — compile-verified
//
#include <hip/hip_runtime.h>

#define N_IN   100000
#define N_OUT  200000
#define C_IN   128
#define C_OUT  64
#define KOFF   27
#define MPAIR  100000
#define BN_EPS 1e-5f

typedef __attribute__((ext_vector_type(2))) float v2f;
typedef __attribute__((ext_vector_type(8))) float v8f;

constexpr int TILES_PER_K       = MPAIR / 16;   // 6250 (exact)
constexpr int WAVES_PER_BLOCK   = 8;            // 256 threads, wave32
constexpr int TILE_BLOCKS_PER_K = (TILES_PER_K + WAVES_PER_BLOCK - 1) / WAVES_PER_BLOCK; // 782

// LDS weight layout: [kpair 0..63][stride 160 floats]: element (kr, n) at
// lds[(kr/2)*160 + n*2 + (kr&1)].  Stride 160 = 32 (mod 64) dwords ->
// lanes 0-15 (kpair p) use banks 0..31, lanes 16-31 (kpair p+1) banks 32..63:
// conflict-free wave32 ds_load_b64.
constexpr int WSTRIDE = 160;                    // floats per kpair row
constexpr int LDSW    = 64 * WSTRIDE;           // 10240 floats = 40 KB

// ---------------------------------------------------------------- zero
__global__ __launch_bounds__(256) void zero_kernel(float4* __restrict__ out4,
                                                   float* __restrict__ stats,
                                                   long long n4) {
  long long i = (long long)blockIdx.x * blockDim.x + threadIdx.x;
  long long stride = (long long)gridDim.x * blockDim.x;
  float4 z = make_float4(0.f, 0.f, 0.f, 0.f);
  for (; i < n4; i += stride) out4[i] = z;
  if (blockIdx.x == 0 && threadIdx.x < 2 * C_OUT) stats[threadIdx.x] = 0.f;
}

// ---------------------------------------------------------------- conv (gather - WMMA - scatter-add)
__global__ __launch_bounds__(256) void conv_kernel(const float* __restrict__ x,
                                                   const float* __restrict__ w,
                                                   const int* __restrict__ in_map,
                                                   const int* __restrict__ out_map,
                                                   float* __restrict__ out) {
  __shared__ float lds_w[LDSW];  // 40 KB: weight[k], K-pair interleaved, all 8 waves share

  const int k  = blockIdx.x / TILE_BLOCKS_PER_K;
  const int tb = blockIdx.x % TILE_BLOCKS_PER_K;

  // cooperative stage of weight[k]: interleave K-pairs so each WMMA B operand
  // is one contiguous 8-byte LDS load. 2048 float2-pairs / 256 thr = 8 iters.
  {
    const float* wk = w + (size_t)k * (C_IN * C_OUT);
    for (int idx = threadIdx.x; idx < 64 * 32; idx += 256) {
      const int kp = idx >> 5;        // kpair 0..63  (rows 2kp, 2kp+1)
      const int n2 = idx & 31;        // column pair 0..31 (cols 2n2, 2n2+1)
      const float2 a = *(const float2*)(wk + (size_t)(2 * kp) * C_OUT + 2 * n2);
      const float2 b = *(const float2*)(wk + (size_t)(2 * kp + 1) * C_OUT + 2 * n2);
      float* d = lds_w + kp * WSTRIDE + 4 * n2;
      *(float2*)(d)     = make_float2(a.x, b.x);   // (2kp, 2n2), (2kp+1, 2n2)
      *(float2*)(d + 2) = make_float2(a.y, b.y);   // (2kp, 2n2+1), (2kp+1, 2n2+1)
    }
  }
  __syncthreads();

  const int wave = threadIdx.x >> 5;
  const int lane = threadIdx.x & 31;
  const int tile = tb * WAVES_PER_BLOCK + wave;
  if (tile >= TILES_PER_K) return;           // wave-uniform: EXEC stays all-ones
  const int m0 = tile * 16;

  const int col   = lane & 15;               // N column within 16-wide tile / A row M
  const int khalf = lane >> 4;               // lanes 16-31 hold K+2 (f32 A/B layout)

  const int* im = in_map  + (size_t)k * MPAIR + m0;
  const int* om = out_map + (size_t)k * MPAIR + m0;

  // A gather: this lane's row of x (row M = lane&15)
  const int row = im[col];
  const float* xr = x + (size_t)row * C_IN + 2 * khalf;

  // B base for this lane: kpair = kk/2 + khalf, column col
  const float* wb = lds_w + (size_t)khalf * WSTRIDE + 2 * col;

  v8f acc0 = {}, acc1 = {}, acc2 = {}, acc3 = {};

  for (int kk = 0; kk < C_IN; kk += 4) {
    // A 16x4 tile: lane holds x[row][kk + 2*khalf + 0..1] (8B-aligned b64 load)
    v2f a = *(const v2f*)(xr + kk);

    // B 4x16 tiles: one ds_load_b64 each, pre-interleaved {K, K+1} pairs
    const float* p = wb + (kk >> 1) * WSTRIDE;
    v2f b0 = *(const v2f*)(p +  0);   // cols n0=0
    v2f b1 = *(const v2f*)(p + 32);   // n0=16
    v2f b2 = *(const v2f*)(p + 64);   // n0=32
    v2f b3 = *(const v2f*)(p + 96);   // n0=48

    acc0 = __builtin_amdgcn_wmma_f32_16x16x4_f32(false, a, false, b0, (short)0, acc0, false, false);
    acc1 = __builtin_amdgcn_wmma_f32_16x16x4_f32(false, a, false, b1, (short)0, acc1, false, false);
    acc2 = __builtin_amdgcn_wmma_f32_16x16x4_f32(false, a, false, b2, (short)0, acc2, false, false);
    acc3 = __builtin_amdgcn_wmma_f32_16x16x4_f32(false, a, false, b3, (short)0, acc3, false, false);
  }

  // scatter-add: C/D layout -> lane owns rows rbase..rbase+7 at column (n0+col)
  const int rbase = khalf * 8;
#pragma unroll
  for (int r = 0; r < 8; ++r) {
    const int orow = om[rbase + r];
    float* dst = out + (size_t)orow * C_OUT + col;
    unsafeAtomicAdd(dst +  0, acc0[r]);
    unsafeAtomicAdd(dst + 16, acc1[r]);
    unsafeAtomicAdd(dst + 32, acc2[r]);
    unsafeAtomicAdd(dst + 48, acc3[r]);
  }
}

// ---------------------------------------------------------------- BN stats (sum, sum^2 per column)
__global__ __launch_bounds__(256) void stats_kernel(const float* __restrict__ out,
                                                    float* __restrict__ stats) {
  const int col  = threadIdx.x & 63;
  const int rsub = threadIdx.x >> 6;   // 4 row-slices per block
  float s = 0.f, s2 = 0.f;
  for (long long r = (long long)blockIdx.x * 4 + rsub; r < N_OUT;
       r += (long long)gridDim.x * 4) {
    float v = out[r * C_OUT + col];
    s  += v;
    s2 += v * v;
  }
  __shared__ float red[2][4][C_OUT];
  red[0][rsub][col] = s;
  red[1][rsub][col] = s2;
  __syncthreads();
  if (rsub == 0) {
    s  = red[0][0][col] + red[0][1][col] + red[0][2][col] + red[0][3][col];
    s2 = red[1][0][col] + red[1][1][col] + red[1][2][col] + red[1][3][col];
    unsafeAtomicAdd(&stats[col], s);
    unsafeAtomicAdd(&stats[C_OUT + col], s2);
  }
}

// ---------------------------------------------------------------- BN apply + ReLU (in place)
__global__ __launch_bounds__(256) void bn_relu_kernel(float* __restrict__ out,
                                                      const float* __restrict__ stats,
                                                      const float* __restrict__ gamma,
                                                      const float* __restrict__ beta) {
  const float inv_n = 1.0f / (float)N_OUT;
  long long i = (long long)blockIdx.x * blockDim.x + threadIdx.x;
  long long stride = (long long)gridDim.x * blockDim.x;
  const long long total = (long long)N_OUT * C_OUT;
  for (; i < total; i += stride) {
    int col = (int)(i & (C_OUT - 1));
    float mean = stats[col] * inv_n;
    float var  = fmaxf(stats[C_OUT + col] * inv_n - mean * mean, 0.f);
    float scale = gamma[col] * rsqrtf(var + BN_EPS);
    float v = (out[i] - mean) * scale + beta[col];
    out[i] = fmaxf(v, 0.f);
  }
}

// ---------------------------------------------------------------- launch
extern "C" void kernel_launch(void* const* d_in, const int* in_sizes, int n_in,
                              void* d_out, int out_size, void* d_ws, size_t ws_size,
                              hipStream_t stream) {
  const float* x      = (const float*)d_in[0];
  const float* weight = (const float*)d_in[1];
  const float* gamma  = (const float*)d_in[2];
  const float* beta   = (const float*)d_in[3];
  const int*   in_map  = (const int*)d_in[4];
  const int*   out_map = (const int*)d_in[5];

  float* out   = (float*)d_out;              // N_OUT * C_OUT f32 accumulator + final result
  float* stats = (float*)d_ws;               // 2 * C_OUT f32: [sum | sumsq]

  const long long n4 = (long long)N_OUT * C_OUT / 4;
  zero_kernel<<<2048, 256, 0, stream>>>((float4*)out, stats, n4);

  const int conv_blocks = KOFF * TILE_BLOCKS_PER_K;  // 27 * 782
  conv_kernel<<<conv_blocks, 256, 0, stream>>>(x, weight, in_map, out_map, out);

  stats_kernel<<<512, 256, 0, stream>>>(out, stats);

  bn_relu_kernel<<<4096, 256, 0, stream>>>(out, stats, gamma, beta);
}